// VQC_25142738551334
// MI455X (gfx1250) — compile-verified
//
#include <hip/hip_runtime.h>
#include <hip/hip_bf16.h>

typedef __attribute__((ext_vector_type(2))) float v2f;
typedef __attribute__((ext_vector_type(8))) float v8f;

static __device__ __forceinline__ v8f wmma_f32_16x16x4(v2f a, v2f b, v8f c) {
  // D = A(16x4) * B(4x16) + C, full fp32
  return __builtin_amdgcn_wmma_f32_16x16x4_f32(false, a, false, b, (short)0, c, false, false);
}

// ---------------------------------------------------------------------------
// Setup: build V_re (32x32), V_im (32x32), Mpad (16x32) in d_ws.
//   V[:,j] = (-i)^pop(j) * ( G_layer1 .. G_layer0 )[:,j]   (post-RX circuit)
//   Mpad[j,i] = sum_q fc_w[j,q] * SIGNS[i,q]   (rows 8..15 zero)
// One wave; lane t owns column t (kept in LDS so all indexing is trivial).
// ---------------------------------------------------------------------------
__global__ __launch_bounds__(32) void vqc_setup(const float* __restrict__ w,
                                                const float* __restrict__ fcw,
                                                float* __restrict__ ws) {
  __shared__ float ur[32][32];
  __shared__ float ui[32][32];
  const int t = threadIdx.x;
  #pragma unroll
  for (int i = 0; i < 32; ++i) { ur[i][t] = (i == t) ? 1.0f : 0.0f; ui[i][t] = 0.0f; }

  #pragma unroll
  for (int l = 0; l < 2; ++l) {
    #pragma unroll
    for (int q = 0; q < 5; ++q) {
      const float phi = w[(l*5 + q)*3 + 0];
      const float th  = w[(l*5 + q)*3 + 1];
      const float om  = w[(l*5 + q)*3 + 2];
      const float c2 = cosf(0.5f*th), s2 = sinf(0.5f*th);
      const float ap = 0.5f*(phi + om), am = 0.5f*(phi - om);
      const float cap = cosf(ap), sap = sinf(ap);
      const float cam = cosf(am), sam = sinf(am);
      // Rot = [[e^{-i ap} c, -e^{i am} s], [e^{-i am} s, e^{i ap} c]]
      const float g00r =  cap*c2, g00i = -sap*c2;
      const float g01r = -cam*s2, g01i = -sam*s2;
      const float g10r =  cam*s2, g10i = -sam*s2;
      const float g11r =  cap*c2, g11i =  sap*c2;
      const int str = 1 << (4 - q);
      #pragma unroll
      for (int i0 = 0; i0 < 32; ++i0) {
        if (i0 & str) continue;
        const int i1 = i0 | str;
        const float xr = ur[i0][t], xi = ui[i0][t];
        const float yr = ur[i1][t], yi = ui[i1][t];
        ur[i0][t] = g00r*xr - g00i*xi + g01r*yr - g01i*yi;
        ui[i0][t] = g00r*xi + g00i*xr + g01r*yi + g01i*yr;
        ur[i1][t] = g10r*xr - g10i*xi + g11r*yr - g11i*yi;
        ui[i1][t] = g10r*xi + g10i*xr + g11r*yi + g11i*yr;
      }
    }
    const int r = (l % 4) + 1;             // CNOT ring range for this layer
    #pragma unroll
    for (int q = 0; q < 5; ++q) {
      const int pc = 4 - q;
      const int pt = 4 - ((q + r) % 5);
      float tr[32], ti[32];
      #pragma unroll
      for (int i = 0; i < 32; ++i) {
        const int pi = i ^ (((i >> pc) & 1) << pt);
        tr[i] = ur[pi][t]; ti[i] = ui[pi][t];
      }
      #pragma unroll
      for (int i = 0; i < 32; ++i) { ur[i][t] = tr[i]; ui[i][t] = ti[i]; }
    }
  }

  // fold (-i)^popcount(column) into the column, store row-major V[i*32 + j]
  const int p4 = __popc(t) & 3;
  #pragma unroll
  for (int i = 0; i < 32; ++i) {
    const float r0 = ur[i][t], m0 = ui[i][t];
    float rn, in_;
    if      (p4 == 0) { rn =  r0; in_ =  m0; }
    else if (p4 == 1) { rn =  m0; in_ = -r0; }
    else if (p4 == 2) { rn = -r0; in_ = -m0; }
    else              { rn = -m0; in_ =  r0; }
    ws[i*32 + t]        = rn;
    ws[1024 + i*32 + t] = in_;
  }

  // Mpad[j, i=t]
  #pragma unroll
  for (int j = 0; j < 16; ++j) {
    float v = 0.0f;
    if (j < 8) {
      #pragma unroll
      for (int q = 0; q < 5; ++q) {
        const float sg = ((t >> (4 - q)) & 1) ? -1.0f : 1.0f;
        v += fcw[j*5 + q] * sg;
      }
    }
    ws[2048 + j*32 + t] = v;
  }
}

// ---------------------------------------------------------------------------
// Main: per 16-row batch tile (one wave):
//   m[b,k] = prod_q (bit(k,q) ? sin : cos)(x[b,q]/2)      (A fragments, analytic)
//   S_re = m * V_re^T, S_im = m * V_im^T   (32 x v_wmma_f32_16x16x4_f32)
//   probs = S_re^2 + S_im^2  -> LDS relayout -> out^T = Mpad * probs^T (8 WMMA)
// ---------------------------------------------------------------------------
__global__ __launch_bounds__(256) void vqc_main(const float* __restrict__ x,
                                                const float* __restrict__ fcb,
                                                const float* __restrict__ ws,
                                                float* __restrict__ out,
                                                int nTiles) {
  __shared__ float Vre_s[1024];
  __shared__ float Vim_s[1024];
  __shared__ float Ms[512];
  __shared__ float Ps[8][16*34 + 2];      // per-wave probs staging, stride 34

  const int tid = threadIdx.x;
  for (int i = tid; i < 1024; i += 256) { Vre_s[i] = ws[i]; Vim_s[i] = ws[1024 + i]; }
  for (int i = tid; i < 512;  i += 256) { Ms[i] = ws[2048 + i]; }
  __syncthreads();

  const int lane = tid & 31;
  const int wave = tid >> 5;
  const int hi   = lane >> 4;             // half-wave: K sub-row select
  const int n0   = lane & 15;

  // Register-resident B fragments of V_re/V_im and A fragments of Mpad.
  // B(4x16) lane layout: N = lane%16, K = kbase + 2*(lane/16) + v
  // A(16x4) lane layout: M = lane%16, K = kbase + 2*(lane/16) + v
  v2f Bre[8][2], Bim[8][2], Am[8];
  #pragma unroll
  for (int kc = 0; kc < 8; ++kc) {
    const int k0 = kc*4 + 2*hi;
    #pragma unroll
    for (int nt = 0; nt < 2; ++nt) {
      const int st = n0 + 16*nt;
      Bre[kc][nt] = *(const v2f*)&Vre_s[st*32 + k0];
      Bim[kc][nt] = *(const v2f*)&Vim_s[st*32 + k0];
    }
    Am[kc] = *(const v2f*)&Ms[n0*32 + k0];
  }

  float fb[8];
  #pragma unroll
  for (int r = 0; r < 8; ++r) fb[r] = fcb[r];

  float* Pw = Ps[wave];
  const v8f vzero = {};

  const int gw     = blockIdx.x * 8 + wave;
  const int stride = gridDim.x * 8;
  for (int tile = gw; tile < nTiles; tile += stride) {
    const int b = tile*16 + n0;
    float c[5], s[5];
    #pragma unroll
    for (int q = 0; q < 5; ++q) {
      const float xv = 0.5f * x[(size_t)b*5 + q];
      c[q] = __cosf(xv);
      s[q] = __sinf(xv);
    }
    // bit(k, q): q0<->k bit4 ... q4<->k bit0.  k = 4*kc + 2*hi + v
    float p[8];
    #pragma unroll
    for (int j = 0; j < 8; ++j)
      p[j] = ((j & 4) ? s[0] : c[0]) * ((j & 2) ? s[1] : c[1]) * ((j & 1) ? s[2] : c[2]);
    const float t3 = hi ? s[3] : c[3];
    const float pc = t3 * c[4];
    const float ps = t3 * s[4];

    v8f aRe[2] = {vzero, vzero};
    v8f aIm[2] = {vzero, vzero};
    #pragma unroll
    for (int kc = 0; kc < 8; ++kc) {
      v2f a; a.x = p[kc] * pc; a.y = p[kc] * ps;
      aRe[0] = wmma_f32_16x16x4(a, Bre[kc][0], aRe[0]);
      aIm[0] = wmma_f32_16x16x4(a, Bim[kc][0], aIm[0]);
      aRe[1] = wmma_f32_16x16x4(a, Bre[kc][1], aRe[1]);
      aIm[1] = wmma_f32_16x16x4(a, Bim[kc][1], aIm[1]);
    }

    // probs -> per-wave LDS, plain [batchrow][state] layout (stride 34)
    #pragma unroll
    for (int nt = 0; nt < 2; ++nt) {
      #pragma unroll
      for (int e = 0; e < 8; ++e) {
        const float pv = aRe[nt][e]*aRe[nt][e] + aIm[nt][e]*aIm[nt][e];
        Pw[(e + 8*hi)*34 + n0 + 16*nt] = pv;
      }
    }

    // out^T(16x16) = Mpad(16x32) * probs^T(32x16)
    v8f aO = vzero;
    #pragma unroll
    for (int kc = 0; kc < 8; ++kc) {
      const int k0 = kc*4 + 2*hi;
      v2f bp = *(const v2f*)&Pw[n0*34 + k0];
      aO = wmma_f32_16x16x4(Am[kc], bp, aO);
    }

    if (hi == 0) {  // lanes 0..15 hold valid output rows j=0..7 (vgpr e), batch n0
      float4 o0, o1;
      o0.x = aO[0] + fb[0]; o0.y = aO[1] + fb[1]; o0.z = aO[2] + fb[2]; o0.w = aO[3] + fb[3];
      o1.x = aO[4] + fb[4]; o1.y = aO[5] + fb[5]; o1.z = aO[6] + fb[6]; o1.w = aO[7] + fb[7];
      float4* op = (float4*)(out + (size_t)(tile*16 + n0) * 8);
      op[0] = o0;
      op[1] = o1;
    }
  }
}

extern "C" void kernel_launch(void* const* d_in, const int* in_sizes, int n_in,
                              void* d_out, int out_size, void* d_ws, size_t ws_size,
                              hipStream_t stream) {
  const float* x   = (const float*)d_in[0];
  const float* w   = (const float*)d_in[1];
  const float* fcw = (const float*)d_in[2];
  const float* fcb = (const float*)d_in[3];
  float* ws  = (float*)d_ws;
  float* out = (float*)d_out;

  const int B      = in_sizes[0] / 5;
  const int nTiles = B / 16;

  vqc_setup<<<1, 32, 0, stream>>>(w, fcw, ws);

  int blocks = (nTiles + 15) / 16;   // 8 waves/block, ~2 tiles per wave
  if (blocks > 1024) blocks = 1024;
  if (blocks < 1)    blocks = 1;
  vqc_main<<<blocks, 256, 0, stream>>>(x, fcb, ws, out, nTiles);
}